// NodeEncoding_72816875537095
// MI455X (gfx1250) — compile-verified
//
#include <hip/hip_runtime.h>
#include <hip/hip_bf16.h>

// B=64 graphs, MAX_A=96, NODE_DIM=256
#define NGRAPH  64
#define MAXA    96
#define NDIM    256
#define ROWS_PER_GRAPH (MAXA * MAXA)        // 9216 (i,j) rows
#define TILES_PER_GRAPH (ROWS_PER_GRAPH/16) // 576
#define EPSV    1e-8f

typedef float v2f __attribute__((ext_vector_type(2)));
typedef float v4f __attribute__((ext_vector_type(4)));
typedef float v8f __attribute__((ext_vector_type(8)));

// ---------------- Kernel 1: zero padded score buffer ----------------
__global__ void ne_zero_sp(float* __restrict__ sp, int n) {
    int i = blockIdx.x * blockDim.x + threadIdx.x;
    if (i < n) sp[i] = 0.0f;
}

// ---------------- Kernel 2: scores = x @ W + b, scatter by pad_idx ----------------
// one wave32 per node row
__global__ __launch_bounds__(256) void ne_scores(const float* __restrict__ x,
                                                 const float* __restrict__ W,
                                                 const float* __restrict__ b,
                                                 const int*   __restrict__ pad_idx,
                                                 float* __restrict__ sp,
                                                 int total) {
    int row  = blockIdx.x * 8 + (threadIdx.x >> 5);
    int lane = threadIdx.x & 31;
    if (row >= total) return;                 // wave-uniform exit
    const float* xr = x + (size_t)row * NDIM;
    float acc = 0.0f;
#pragma unroll
    for (int i = 0; i < NDIM / 32; ++i) {
        int idx = lane + 32 * i;
        acc += xr[idx] * W[idx];
    }
#pragma unroll
    for (int off = 16; off > 0; off >>= 1)
        acc += __shfl_xor(acc, off, 32);
    if (lane == 0) sp[pad_idx[row]] = acc + b[0];
}

// ---------------- Kernel 3: num/den via V_WMMA_F32_16X16X4_F32 ----------------
// A tile  = paths[b, rowTile(16), k..k+3]   (16x4, f32)
// B tile  = [ sp[b,k..k+3] | ones | 0 ... ] (4x16, f32)  -> col0=num, col1=den
// Each wave handles one 16-row tile across all k (24 WMMA issues).
__global__ __launch_bounds__(256) void ne_path_avg(const float* __restrict__ paths,
                                                   const float* __restrict__ sp,
                                                   float* __restrict__ out) {
    __shared__ float sp_lds[MAXA];
    const int b = blockIdx.y;
    if (threadIdx.x < MAXA)
        sp_lds[threadIdx.x] = sp[b * MAXA + threadIdx.x];
    __syncthreads();

    const int wave = threadIdx.x >> 5;
    const int lane = threadIdx.x & 31;
    const int half = lane >> 4;     // 0: K pair {0,1}, 1: K pair {2,3}
    const int n    = lane & 15;     // A: row-in-tile (M) ; B/C/D: column (N)

    const int tile     = blockIdx.x * 8 + wave;   // 0..575
    const int row_base = tile * 16;

    // A-operand pointer for this lane: row (row_base+n), starting k = 2*half
    const float* __restrict__ prow =
        paths + (((size_t)b * ROWS_PER_GRAPH) + row_base + n) * MAXA + 2 * half;
    __builtin_prefetch(prow, 0, 1);

    v8f c = {};
#pragma unroll 4
    for (int kb = 0; kb < MAXA; kb += 4) {
        // A: two consecutive k values per lane (8B global load, row-contiguous stream)
        v2f a = *(const v2f*)(prow + kb);
        // B: lane column n gets sp (n==0), ones (n==1), zero otherwise
        v2f s = *(const v2f*)(&sp_lds[kb + 2 * half]);
        v2f bv;
        bv.x = (n == 0) ? s.x : ((n == 1) ? 1.0f : 0.0f);
        bv.y = (n == 0) ? s.y : ((n == 1) ? 1.0f : 0.0f);
        // D = A*B + C  (full f32; emits v_wmma_f32_16x16x4_f32)
        c = __builtin_amdgcn_wmma_f32_16x16x4_f32(
                /*neg_a=*/false, a, /*neg_b=*/false, bv,
                /*c_mod=*/(short)0, c, /*reuse_a=*/false, /*reuse_b=*/false);
    }

    // C/D layout: vgpr g, lanes 0-15 -> M=g (N=lane), lanes 16-31 -> M=8+g
    // num for row m lives at lane (half*16 + 0), den at lane (half*16 + 1).
    // Compute uniformly in all lanes (no divergence), then two packed 128-bit
    // stores from lanes 0 (m=0..7) and 16 (m=8..15).
    float vals[8];
#pragma unroll
    for (int g = 0; g < 8; ++g) {
        float numv = __shfl(c[g], (half << 4) | 0, 32);
        float denv = __shfl(c[g], (half << 4) | 1, 32);
        vals[g] = numv * __builtin_amdgcn_rcpf(denv + EPSV);
    }
    float* __restrict__ orow = out + (size_t)b * ROWS_PER_GRAPH + row_base + 8 * half;
    if (n == 0) {
        v4f lo = { vals[0], vals[1], vals[2], vals[3] };
        v4f hi = { vals[4], vals[5], vals[6], vals[7] };
        *(v4f*)(orow + 0) = lo;   // 16B-aligned (row_base is 64B-aligned)
        *(v4f*)(orow + 4) = hi;
    }
}

extern "C" void kernel_launch(void* const* d_in, const int* in_sizes, int n_in,
                              void* d_out, int out_size, void* d_ws, size_t ws_size,
                              hipStream_t stream) {
    const float* x       = (const float*)d_in[0];
    const float* W       = (const float*)d_in[1];
    const float* bscal   = (const float*)d_in[2];
    const float* paths   = (const float*)d_in[3];
    const int*   pad_idx = (const int*)d_in[4];
    float*       out     = (float*)d_out;
    float*       sp      = (float*)d_ws;          // NGRAPH*MAXA floats = 24 KB

    const int total    = in_sizes[4];             // number of nodes (4395)
    const int sp_elems = NGRAPH * MAXA;           // 6144

    ne_zero_sp<<<(sp_elems + 255) / 256, 256, 0, stream>>>(sp, sp_elems);
    ne_scores<<<(total + 7) / 8, 256, 0, stream>>>(x, W, bscal, pad_idx, sp, total);
    ne_path_avg<<<dim3(TILES_PER_GRAPH / 8, NGRAPH), 256, 0, stream>>>(paths, sp, out);
}